// Fusionloss_LS_32908039422556
// MI455X (gfx1250) — compile-verified
//
#include <hip/hip_runtime.h>

#define HH 384
#define WW 384
#define CC 3
#define BB 4
#define NBOX 10
#define TILE_W 32
#define TILE_H 8
#define HALO_W 34
#define HALO_H 10
#define GX 12   // WW / TILE_W
#define GY 48   // HH / TILE_H
#define BLOCKS_PER_BATCH (3 * GY * GX)   // 1728
#define WS1_COUNT (BB * BLOCKS_PER_BATCH * 3)        // 20736 floats
#define WS2_COUNT (BB * NBOX * GY * 2)               // 3840 floats

typedef __attribute__((ext_vector_type(2))) float v2f;
typedef __attribute__((ext_vector_type(8))) float v8f;

static __device__ __forceinline__ int imin(int a, int b) { return a < b ? a : b; }
static __device__ __forceinline__ int imax(int a, int b) { return a > b ? a : b; }

// Full-wave (32 lane) f32 sum using one V_WMMA_F32_16X16X4_F32.
// A = ones(16x4). B: lane L writes b = {v, 0}; B[0][n]=v[n], B[2][n]=v[n+16],
// B[1]=B[3]=0 -> D[m][n] = v[n] + v[n+16] for every row m.
// c[0] on lane L therefore holds colsum(L%16); 4 xor-shuffles finish the sum.
static __device__ __forceinline__ float wave_reduce_sum(float v) {
    v2f a; a.x = 1.0f; a.y = 1.0f;
    v2f b; b.x = v;    b.y = 0.0f;
    v8f c = {};
    c = __builtin_amdgcn_wmma_f32_16x16x4_f32(
        /*neg_a=*/false, a, /*neg_b=*/false, b,
        /*c_mod=*/(short)0, c, /*reuse_a=*/false, /*reuse_b=*/false);
    float s = c[0];
    s += __shfl_xor(s, 8, 32);
    s += __shfl_xor(s, 4, 32);
    s += __shfl_xor(s, 2, 32);
    s += __shfl_xor(s, 1, 32);
    return s;
}

static __device__ __forceinline__ float sobel_lds(const float a[HALO_H][HALO_W], int r, int c) {
    float tl = a[r - 1][c - 1], tp = a[r - 1][c], tr = a[r - 1][c + 1];
    float lf = a[r][c - 1],                      rg = a[r][c + 1];
    float bl = a[r + 1][c - 1], bt = a[r + 1][c], br = a[r + 1][c + 1];
    float gx = (tr + 2.0f * rg + br) - (tl + 2.0f * lf + bl);
    float gy = (tl + 2.0f * tp + tr) - (bl + 2.0f * bt + br);
    return fabsf(gx) + fabsf(gy);
}

static __device__ __forceinline__ float sobel_reg(const float a[3][3]) {
    float gx = (a[0][2] + 2.0f * a[1][2] + a[2][2]) - (a[0][0] + 2.0f * a[1][0] + a[2][0]);
    float gy = (a[0][0] + 2.0f * a[0][1] + a[0][2]) - (a[2][0] + 2.0f * a[2][1] + a[2][2]);
    return fabsf(gx) + fabsf(gy);
}

// ---------------------------------------------------------------------------
// Kernel 1: dense per-pixel losses with LDS halo tiling.
// grid = (GX, GY, B*C), block = 256 (=32x8). Each block -> one 32x8 tile.
// Writes per-block partials {sum_in, sum_grad, sum_mse} to ws1[bf*3..].
// ---------------------------------------------------------------------------
__global__ __launch_bounds__(256) void global_loss_kernel(
    const float* __restrict__ vis, const float* __restrict__ ir,
    const float* __restrict__ gen, const float* __restrict__ th,
    float* __restrict__ ws1) {
    __shared__ float sv[HALO_H][HALO_W];
    __shared__ float sm[HALO_H][HALO_W];
    __shared__ float sg[HALO_H][HALO_W];
    __shared__ float wred[8][3];

    const int z   = blockIdx.z;                 // b*C + c plane
    const int r0  = blockIdx.y * TILE_H;
    const int c0  = blockIdx.x * TILE_W;
    const int tid = threadIdx.x;

    const size_t plane = (size_t)z * (HH * WW);
    const float* pv = vis + plane;
    const float* pt = th  + plane;
    const float* pg = gen + plane;
    const float* pi = ir  + plane;

    for (int i = tid; i < HALO_H * HALO_W; i += 256) {
        int rr = i / HALO_W, cc = i % HALO_W;
        int gr = r0 - 1 + rr, gc = c0 - 1 + cc;
        float v = 0.0f, t = 0.0f, g = 0.0f;
        if (gr >= 0 && gr < HH && gc >= 0 && gc < WW) {
            int o = gr * WW + gc;
            v = pv[o]; t = pt[o]; g = pg[o];
        }
        sv[rr][cc] = v;
        sm[rr][cc] = 0.5f * v + 0.5f * t;   // x_in_mean (0 outside image = conv zero-pad)
        sg[rr][cc] = g;
    }
    __syncthreads();

    const int lx = tid & 31, ly = tid >> 5;     // 32 x 8 pixel map
    const int rr = ly + 1, cc = lx + 1;

    float vC  = sv[rr][cc];
    float mC  = sm[rr][cc];
    float gC  = sg[rr][cc];
    float iC  = pi[(r0 + ly) * WW + (c0 + lx)];

    float yg = sobel_lds(sv, rr, cc);           // sobel(vis)
    float mg = sobel_lds(sm, rr, cc);           // sobel(x_in_mean)
    float gg = sobel_lds(sg, rr, cc);           // sobel(gen)

    float t_in   = fabsf(gC - fmaxf(vC, mC));
    float t_grad = fabsf(gg - fmaxf(yg, mg));
    float dv = gC - vC, di = gC - iC;
    float t_mse = dv * dv + di * di;

    float s_in = wave_reduce_sum(t_in);
    float s_gr = wave_reduce_sum(t_grad);
    float s_ms = wave_reduce_sum(t_mse);

    if (lx == 0) { wred[ly][0] = s_in; wred[ly][1] = s_gr; wred[ly][2] = s_ms; }
    __syncthreads();
    if (tid == 0) {
        float a0 = 0.0f, a1 = 0.0f, a2 = 0.0f;
        for (int w = 0; w < 8; ++w) { a0 += wred[w][0]; a1 += wred[w][1]; a2 += wred[w][2]; }
        int bf = (blockIdx.z * GY + blockIdx.y) * GX + blockIdx.x;
        ws1[bf * 3 + 0] = a0;
        ws1[bf * 3 + 1] = a1;
        ws1[bf * 3 + 2] = a2;
    }
}

// ---------------------------------------------------------------------------
// Kernel 2: masked box losses. grid = (GY row-chunks, NBOX, B), block = 256.
// Only pixels inside the box contribute (mask multiplier); masked 3x3
// neighborhoods are synthesized on the fly (zeros outside rect / image).
// Writes per-chunk partials {ssim_sum, l1_sum} to ws2. Always writes its slot.
// ---------------------------------------------------------------------------
__global__ __launch_bounds__(256) void box_loss_kernel(
    const float* __restrict__ vis, const float* __restrict__ gen,
    const float* __restrict__ th, const int* __restrict__ label,
    float* __restrict__ ws2) {
    __shared__ float wred[8][2];
    const int b = blockIdx.z, n = blockIdx.y, chunk = blockIdx.x;
    const int tid = threadIdx.x;

    const int* lb = label + (b * NBOX + n) * 5;
    const int x1 = lb[1], y1 = lb[2], x2 = lb[3], y2 = lb[4];
    const bool valid = (x1 | y1 | x2 | y2) != 0;

    const int rA = imax(imax(y1, chunk * TILE_H), 0);
    const int rB = imin(imin(y2, chunk * TILE_H + TILE_H), HH);
    const int cA = imax(x1, 0);
    const int cB = imin(x2, WW);
    const int nr = rB - rA, nc = cB - cA;

    float accS = 0.0f, accL = 0.0f;
    if (valid && nr > 0 && nc > 0) {
        const int per = nr * nc;
        const int total = per * CC;
        for (int idx = tid; idx < total; idx += 256) {
            int ch  = idx / per;
            int rem = idx - ch * per;
            int r = rA + rem / nc;
            int c = cA + rem % nc;
            const size_t plane = (size_t)(b * CC + ch) * (HH * WW);
            const float* pv = vis + plane;
            const float* pt = th  + plane;
            const float* pg = gen + plane;

            float mv[3][3], mm[3][3], mg[3][3];
            #pragma unroll
            for (int dr = -1; dr <= 1; ++dr) {
                #pragma unroll
                for (int dc = -1; dc <= 1; ++dc) {
                    int rrp = r + dr, ccp = c + dc;
                    float vv = 0.0f, tt = 0.0f, gv = 0.0f;
                    if (rrp >= y1 && rrp < y2 && ccp >= x1 && ccp < x2 &&
                        rrp >= 0 && rrp < HH && ccp >= 0 && ccp < WW) {
                        int o = rrp * WW + ccp;
                        vv = pv[o]; tt = pt[o]; gv = pg[o];
                    }
                    mv[dr + 1][dc + 1] = vv;
                    mm[dr + 1][dc + 1] = 0.3f * vv + 0.7f * tt;  // masked x_in_mean_7
                    mg[dr + 1][dc + 1] = gv;
                }
            }
            float gvs = sobel_reg(mv);
            float gms = sobel_reg(mm);
            float ggs = sobel_reg(mg);
            float grad_max = fmaxf(gvs, gms);
            float in_max   = fmaxf(mv[1][1], mm[1][1]);   // mask==1 at center
            accS += fabsf(ggs - grad_max);
            accL += fabsf(mg[1][1] - in_max);
        }
    }

    float sS = wave_reduce_sum(accS);
    float sL = wave_reduce_sum(accL);
    const int lane = tid & 31, wv = tid >> 5;
    if (lane == 0) { wred[wv][0] = sS; wred[wv][1] = sL; }
    __syncthreads();
    if (tid == 0) {
        float a0 = 0.0f, a1 = 0.0f;
        for (int w = 0; w < 8; ++w) { a0 += wred[w][0]; a1 += wred[w][1]; }
        int o = ((b * NBOX + n) * GY + chunk) * 2;
        ws2[o + 0] = a0;
        ws2[o + 1] = a1;
    }
}

// ---------------------------------------------------------------------------
// Kernel 3: finalize. Single block, fixed-order (deterministic) reductions.
// Output layout: [loss_ss(4), loss_global(4), loss_label(4), loss_in(4),
//                 loss_grad(4), ls(4), lin(4)] = 28 floats.
// ---------------------------------------------------------------------------
__global__ __launch_bounds__(256) void finalize_kernel(
    const float* __restrict__ alpha_p, const float* __restrict__ beta_p,
    const int* __restrict__ label, const float* __restrict__ ws1,
    const float* __restrict__ ws2, float* __restrict__ out) {
    __shared__ float red[256];
    __shared__ float gsums[BB][3];
    __shared__ float lssim_s[BB][NBOX];
    __shared__ float l1_s[BB][NBOX];
    const int tid = threadIdx.x;

    // Dense-loss sums per batch (3 accumulators each).
    for (int b = 0; b < BB; ++b) {
        for (int k = 0; k < 3; ++k) {
            float p = 0.0f;
            for (int i = tid; i < BLOCKS_PER_BATCH; i += 256)
                p += ws1[(b * BLOCKS_PER_BATCH + i) * 3 + k];
            red[tid] = p;
            __syncthreads();
            for (int s = 128; s > 0; s >>= 1) {
                if (tid < s) red[tid] += red[tid + s];
                __syncthreads();
            }
            if (tid == 0) gsums[b][k] = red[0];
            __syncthreads();
        }
    }

    // Per-box sums + normalization.
    if (tid < BB * NBOX) {
        int b = tid / NBOX, n = tid % NBOX;
        float ss = 0.0f, l1 = 0.0f;
        for (int j = 0; j < GY; ++j) {
            int o = ((b * NBOX + n) * GY + j) * 2;
            ss += ws2[o + 0];
            l1 += ws2[o + 1];
        }
        const int* lb = label + (b * NBOX + n) * 5;
        int x1 = lb[1], y1 = lb[2], x2 = lb[3], y2 = lb[4];
        bool valid = (x1 | y1 | x2 | y2) != 0;
        int nr = imin(y2, HH) - imax(y1, 0); nr = imax(nr, 0);
        int nc = imin(x2, WW) - imax(x1, 0); nc = imax(nc, 0);
        float count = (float)nr * (float)nc;
        float denom = fmaxf(count * (float)CC, 1.0f);
        lssim_s[b][n] = valid ? ss / denom : 0.0f;
        l1_s[b][n]    = valid ? l1 / denom : 0.0f;
    }
    __syncthreads();

    if (tid < BB) {
        const int b = tid;
        const float alpha = alpha_p[0];
        const float beta  = beta_p[0];
        const float inv_chw = 1.0f / ((float)CC * (float)HH * (float)WW);
        float loss_in   = 0.5f  * gsums[b][0] * inv_chw;
        float loss_grad =         gsums[b][1] * inv_chw;
        float loss_mse  = 0.25f * gsums[b][2] * inv_chw;
        float loss_global = alpha * (loss_in + loss_mse) + (1.0f - alpha) * loss_grad;

        float ex = 0.0f, ssum = 0.0f, lsum = 0.0f;
        for (int n = 0; n < NBOX; ++n) {
            float s = lssim_s[b][n], l = l1_s[b][n];
            if (s != 0.0f || l != 0.0f) ex += 1.0f;
            ssum += s;
            lsum += l;
        }
        float safe = fmaxf(ex, 1.0f);
        float ls  = (ex > 0.0f) ? ssum / safe : 0.0f;
        float lin = (ex > 0.0f) ? lsum / safe : 0.0f;
        float loss_label = (ex > 0.0f) ? (1.0f - beta) * ls + beta * lin : 0.0f;

        out[0  + b] = 0.0f;          // loss_ss
        out[4  + b] = loss_global;
        out[8  + b] = loss_label;
        out[12 + b] = loss_in;
        out[16 + b] = loss_grad;
        out[20 + b] = ls;
        out[24 + b] = lin;
    }
}

extern "C" void kernel_launch(void* const* d_in, const int* in_sizes, int n_in,
                              void* d_out, int out_size, void* d_ws, size_t ws_size,
                              hipStream_t stream) {
    (void)in_sizes; (void)n_in; (void)out_size; (void)ws_size;
    const float* alpha = (const float*)d_in[0];
    const float* beta  = (const float*)d_in[1];
    const float* vis   = (const float*)d_in[2];
    const float* ir    = (const float*)d_in[3];
    const float* gen   = (const float*)d_in[4];
    const int*   label = (const int*)d_in[5];
    const float* th    = (const float*)d_in[6];

    float* ws1 = (float*)d_ws;              // WS1_COUNT floats
    float* ws2 = ws1 + WS1_COUNT;           // WS2_COUNT floats

    dim3 g1(GX, GY, BB * CC);
    global_loss_kernel<<<g1, 256, 0, stream>>>(vis, ir, gen, th, ws1);

    dim3 g2(GY, NBOX, BB);
    box_loss_kernel<<<g2, 256, 0, stream>>>(vis, gen, th, label, ws2);

    finalize_kernel<<<1, 256, 0, stream>>>(alpha, beta, label, ws1, ws2, (float*)d_out);
}